// LlamaForCausalLM_57878979281335
// MI455X (gfx1250) — compile-verified
//
#include <hip/hip_runtime.h>

typedef __attribute__((ext_vector_type(2))) float v2f;
typedef __attribute__((ext_vector_type(8))) float v8f;

#define HDIM  2048
#define NHEAD 16
#define NKVH  8
#define HD    128
#define IDIM  5632
#define NLAYER 2
#define VOCAB 32000
#define BBATCH 2
#define SLEN  2048
#define NTOK  (BBATCH * SLEN)
#define QKV_N ((NHEAD + 2 * NKVH) * HD)   // 4096

// async global -> LDS copy of 16 bytes (per lane), tracked by ASYNCcnt
__device__ __forceinline__ void async_copy_b128(const float* gptr, float* lptr) {
  unsigned lds = (unsigned)(size_t)lptr;          // low 32 bits of generic ptr == LDS offset
  unsigned long long ga = (unsigned long long)(size_t)gptr;
  asm volatile("global_load_async_to_lds_b128 %0, %1, off" :: "v"(lds), "v"(ga) : "memory");
}
__device__ __forceinline__ void wait_async0() {
  asm volatile("s_wait_asynccnt 0x0" ::: "memory");
}

// ---------------------------------------------------------------- embed gather
__global__ void embed_kernel(const int* __restrict__ ids,
                             const float* __restrict__ embed,
                             float* __restrict__ h) {
  int tok = blockIdx.x;
  int id  = ids[tok];
  const float4* src = (const float4*)(embed + (size_t)id * HDIM);
  float4*       dst = (float4*)(h + (size_t)tok * HDIM);
  for (int i = threadIdx.x; i < HDIM / 4; i += blockDim.x) dst[i] = src[i];
}

// ---------------------------------------------------------------- rmsnorm (256 thr / token)
__global__ __launch_bounds__(256) void rmsnorm_kernel(const float* __restrict__ x,
                                                      const float* __restrict__ w,
                                                      float* __restrict__ y) {
  int tok = blockIdx.x;
  const float* xr = x + (size_t)tok * HDIM;
  float vals[8];
  float ss = 0.f;
  for (int i = 0; i < 8; i++) {
    vals[i] = xr[threadIdx.x + i * 256];
    ss += vals[i] * vals[i];
  }
  for (int m = 16; m >= 1; m >>= 1) ss += __shfl_xor(ss, m, 32);
  __shared__ float red[8];
  if ((threadIdx.x & 31) == 0) red[threadIdx.x >> 5] = ss;
  __syncthreads();
  float tot = 0.f;
  for (int i = 0; i < 8; i++) tot += red[i];
  float inv = rsqrtf(tot / (float)HDIM + 1e-5f);
  float* yr = y + (size_t)tok * HDIM;
  for (int i = 0; i < 8; i++) {
    int c = threadIdx.x + i * 256;
    yr[c] = vals[i] * inv * w[c];
  }
}

// ---------------------------------------------------------------- fp32 WMMA GEMM
// C[M,N] = A[M,K] @ W[K,N] (+ R).  Block tile 128x128, 8 waves, BK=32.
// Double-buffered LDS; A staged by global_load_async_to_lds_b128 (ASYNCcnt),
// B staged via VGPRs (k-pair interleave) with loads issued before compute.
// B LDS layout: element (k,n) at [k/2][n*2+(k&1)] -> B fragment = one ds_load_b64.
#define BM 128
#define BN 128
#define BK 32
#define AS_STR 36          // padded A stride (floats): conflict-free, rows 16B aligned
#define BS_STR 288         // pair-row stride: 2*BN+32, (288 % 64)==32 -> halves disjoint

__global__ __launch_bounds__(256) void gemm_kernel(const float* __restrict__ A,
                                                   const float* __restrict__ W,
                                                   const float* __restrict__ R,
                                                   float* __restrict__ C,
                                                   int M, int N, int K) {
  __shared__ __align__(16) float As[2][BM * AS_STR];
  __shared__ __align__(16) float Bs[2][(BK / 2) * BS_STR];
  int n0 = blockIdx.x * BN;
  int m0 = blockIdx.y * BM;
  int tid  = threadIdx.x;
  int wv   = tid >> 5;
  int lane = tid & 31;
  int ln   = lane & 15;
  int koff = (lane < 16) ? 0 : 2;
  int hi   = (lane < 16) ? 0 : 8;

  v8f acc[8];
  for (int t = 0; t < 8; t++)
    for (int e = 0; e < 8; e++) acc[t][e] = 0.f;

  float2 b0[4], b1[4];

  // issue async A-tile copy (128x32) into As[buf]
  auto issueA = [&](int k0, int buf) {
    for (int i = 0; i < 4; i++) {
      int idx = tid + i * 256;            // 1024 float4s
      int row = idx >> 3, c4 = idx & 7;
      async_copy_b128(A + (size_t)(m0 + row) * K + k0 + c4 * 4,
                      &As[buf][row * AS_STR + c4 * 4]);
    }
  };
  // issue global loads for B-tile (32x128) into registers
  auto issueB = [&](int k0) {
    for (int i = 0; i < 4; i++) {
      int idx = tid + i * 256;            // 1024 tasks: 16 pairs x 64 float2-cols
      int pr = idx >> 6, c2 = idx & 63;
      const float* g0 = W + (size_t)(k0 + pr * 2) * N + n0 + c2 * 2;
      b0[i] = *(const float2*)g0;
      b1[i] = *(const float2*)(g0 + N);
    }
  };
  // pack + store B registers into Bs[buf] (k-pair interleaved)
  auto storeB = [&](int buf) {
    for (int i = 0; i < 4; i++) {
      int idx = tid + i * 256;
      int pr = idx >> 6, c2 = idx & 63;
      float4 st; st.x = b0[i].x; st.y = b1[i].x; st.z = b0[i].y; st.w = b1[i].y;
      *(float4*)(&Bs[buf][pr * BS_STR + c2 * 4]) = st;
    }
  };

  // prologue: stage first tiles
  issueA(0, 0);
  issueB(0);
  storeB(0);
  wait_async0();
  __syncthreads();

  int buf = 0;
  for (int k0 = 0; k0 < K; k0 += BK) {
    bool nxt = (k0 + BK) < K;
    if (nxt) {                 // overlap next-tile global traffic with compute
      issueA(k0 + BK, buf ^ 1);
      issueB(k0 + BK);
    }

    const float* Arow = &As[buf][(wv * 16 + ln) * AS_STR];
    const float* Bbuf = &Bs[buf][0];
    for (int s = 0; s < 8; s++) {
      int kk = s * 4 + koff;
      v2f a = *(const v2f*)(Arow + kk);
      const float* Bp = Bbuf + (kk >> 1) * BS_STR + ln * 2;
      for (int t = 0; t < 8; t++) {
        v2f b = *(const v2f*)(Bp + t * 32);
        acc[t] = __builtin_amdgcn_wmma_f32_16x16x4_f32(false, a, false, b,
                                                       (short)0, acc[t], false, false);
      }
    }

    if (nxt) storeB(buf ^ 1);  // loadcnt wait lands here, behind 64 WMMAs
    wait_async0();
    __syncthreads();
    buf ^= 1;
  }

  int mrow = m0 + wv * 16;
  for (int t = 0; t < 8; t++)
    for (int r = 0; r < 8; r++) {
      int row = mrow + r + hi;
      int col = n0 + t * 16 + ln;
      float v = acc[t][r];
      if (R) v += R[(size_t)row * N + col];
      C[(size_t)row * N + col] = v;
    }
}

// ---------------------------------------------------------------- RoPE + q/k/v split
__global__ __launch_bounds__(256) void rope_kernel(const float* __restrict__ qkv,
                                                   const int* __restrict__ positions,
                                                   float* __restrict__ Q,
                                                   float* __restrict__ Kc,
                                                   float* __restrict__ Vc) {
  int tok = blockIdx.x;
  int b = tok / SLEN, s = tok % SLEN;
  float pos = (float)positions[tok];
  const float* src = qkv + (size_t)tok * QKV_N;
  const float kln = 9.210340371976184f / 128.0f;  // ln(10000)/128

  for (int idx = threadIdx.x; idx < NHEAD * 64; idx += blockDim.x) {
    int hh = idx >> 6, d = idx & 63;
    float ang = pos * __expf(-(float)(2 * d) * kln);
    float cs = __cosf(ang), sn = __sinf(ang);
    float x1 = src[hh * HD + d], x2 = src[hh * HD + d + 64];
    float* dst = Q + (((size_t)b * NHEAD + hh) * SLEN + s) * HD;
    dst[d]      = x1 * cs - x2 * sn;
    dst[d + 64] = x2 * cs + x1 * sn;
  }
  for (int idx = threadIdx.x; idx < NKVH * 64; idx += blockDim.x) {
    int hh = idx >> 6, d = idx & 63;
    float ang = pos * __expf(-(float)(2 * d) * kln);
    float cs = __cosf(ang), sn = __sinf(ang);
    const float* ks = src + NHEAD * HD;
    float x1 = ks[hh * HD + d], x2 = ks[hh * HD + d + 64];
    float* dst = Kc + (((size_t)b * NKVH + hh) * SLEN + s) * HD;
    dst[d]      = x1 * cs - x2 * sn;
    dst[d + 64] = x2 * cs + x1 * sn;
  }
  for (int idx = threadIdx.x; idx < NKVH * HD; idx += blockDim.x) {
    int hh = idx >> 7, d = idx & 127;
    Vc[(((size_t)b * NKVH + hh) * SLEN + s) * HD + d] = src[(NHEAD + NKVH) * HD + hh * HD + d];
  }
}

// ---------------------------------------------------------------- WMMA flash attention
// grid (B*NH, S/64); 4 waves, each owns 16 q rows; online softmax over 16-key blocks.
#define QTILE 64
#define VS_STR 288   // V pair-row stride: 2*HD+32

__global__ __launch_bounds__(128) void attn_kernel(const float* __restrict__ Q,
                                                   const float* __restrict__ K,
                                                   const float* __restrict__ V,
                                                   float* __restrict__ O) {
  __shared__ __align__(16) float Qs[QTILE * 132];
  __shared__ __align__(16) float Ks[16 * 132];
  __shared__ __align__(16) float Vs[8 * VS_STR];   // k-pair interleaved
  __shared__ __align__(16) float Ps[4][16 * 20];

  int bh = blockIdx.x;
  int b = bh / NHEAD, hq = bh % NHEAD;
  int kvh = hq / (NHEAD / NKVH);
  int q0 = blockIdx.y * QTILE;
  int tid  = threadIdx.x;
  int wv   = tid >> 5;
  int lane = tid & 31;
  int ln   = lane & 15;
  int koff = (lane < 16) ? 0 : 2;
  int hi   = (lane < 16) ? 0 : 8;
  const float scale = 0.08838834764831845f;  // 1/sqrt(128)

  // async-stage Q tile (64 x 128), waited on before first compute
  const float* Qbase = Q + (((size_t)b * NHEAD + hq) * SLEN + q0) * HD;
  for (int i = 0; i < 16; i++) {
    int idx = tid + i * 128;                // 2048 float4s
    int row = idx >> 5, c4 = idx & 31;
    async_copy_b128(Qbase + (size_t)row * HD + c4 * 4, &Qs[row * 132 + c4 * 4]);
  }

  v8f o[8];
  float m_run[8], l_run[8];
  for (int t = 0; t < 8; t++)
    for (int e = 0; e < 8; e++) o[t][e] = 0.f;
  for (int r = 0; r < 8; r++) { m_run[r] = -1e30f; l_run[r] = 0.f; }

  const float* Kbase = K + ((size_t)b * NKVH + kvh) * SLEN * HD;
  const float* Vbase = V + ((size_t)b * NKVH + kvh) * SLEN * HD;
  int jmax = (q0 + QTILE - 1) >> 4;

  for (int j = 0; j <= jmax; j++) {
    __syncthreads();
    // K block: async copy, row-major (16 x 128, padded stride 132)
    for (int i = 0; i < 4; i++) {
      int idx = tid + i * 128;              // 512 float4s
      int row = idx >> 5, c4 = idx & 31;
      async_copy_b128(Kbase + (size_t)(j * 16 + row) * HD + c4 * 4,
                      &Ks[row * 132 + c4 * 4]);
    }
    // V block: k-pair interleaved via VGPRs (8 pairs x 256 floats)
    for (int i = 0; i < 4; i++) {
      int idx = tid + i * 128;              // 512 tasks: 8 pairs x 64 float2-cols
      int pr = idx >> 6, c2 = idx & 63;
      const float* g0 = Vbase + (size_t)(j * 16 + pr * 2) * HD + c2 * 2;
      float2 v0 = *(const float2*)g0;
      float2 v1 = *(const float2*)(g0 + HD);
      float4 st; st.x = v0.x; st.y = v1.x; st.z = v0.y; st.w = v1.y;
      *(float4*)(&Vs[pr * VS_STR + c2 * 4]) = st;
    }
    wait_async0();
    __syncthreads();

    // S = Q(16x128) . K^T(128x16) : 32 WMMAs, K=4 each
    v8f sc;
    for (int e = 0; e < 8; e++) sc[e] = 0.f;
    const float* Qrow = Qs + (wv * 16 + ln) * 132;
    const float* Krow = Ks + ln * 132;
    for (int s = 0; s < 32; s++) {
      int kk = s * 4 + koff;
      v2f a  = *(const v2f*)(Qrow + kk);
      v2f bf = *(const v2f*)(Krow + kk);
      sc = __builtin_amdgcn_wmma_f32_16x16x4_f32(false, a, false, bf,
                                                 (short)0, sc, false, false);
    }

    // online softmax per q-row (rows 0-7 live in lanes 0-15, rows 8-15 in lanes 16-31)
    for (int r = 0; r < 8; r++) {
      float sv = sc[r] * scale;
      int gq  = q0 + wv * 16 + r + hi;
      int key = j * 16 + ln;
      if (key > gq) sv = -1e30f;
      float mt = sv;
      for (int msk = 8; msk >= 1; msk >>= 1) mt = fmaxf(mt, __shfl_xor(mt, msk, 16));
      float mnew = fmaxf(m_run[r], mt);
      float corr = __expf(m_run[r] - mnew);
      float p = __expf(sv - mnew);
      float psum = p;
      for (int msk = 8; msk >= 1; msk >>= 1) psum += __shfl_xor(psum, msk, 16);
      l_run[r] = l_run[r] * corr + psum;
      m_run[r] = mnew;
      for (int t = 0; t < 8; t++) o[t][r] *= corr;
      Ps[wv][(r + hi) * 20 + ln] = p;   // C-layout -> LDS for A-layout reload
    }
    __syncthreads();

    // O += P(16x16) . V(16x128) : 4 K-steps x 8 d-tiles
    for (int s = 0; s < 4; s++) {
      int kk = s * 4 + koff;
      v2f a = *(const v2f*)(&Ps[wv][ln * 20 + kk]);
      const float* Vp = Vs + (kk >> 1) * VS_STR + ln * 2;
      for (int t = 0; t < 8; t++) {
        v2f bf = *(const v2f*)(Vp + t * 32);
        o[t] = __builtin_amdgcn_wmma_f32_16x16x4_f32(false, a, false, bf,
                                                     (short)0, o[t], false, false);
      }
    }
  }

  for (int t = 0; t < 8; t++)
    for (int r = 0; r < 8; r++) {
      int q = q0 + wv * 16 + r + hi;
      float val = o[t][r] / l_run[r];
      O[((size_t)(b * SLEN + q)) * (NHEAD * HD) + hq * HD + t * 16 + ln] = val;
    }
}

// ---------------------------------------------------------------- silu(gate)*up
__global__ void silu_mul_kernel(const float* __restrict__ gu, float* __restrict__ ff) {
  size_t i = (size_t)blockIdx.x * blockDim.x + threadIdx.x;
  if (i >= (size_t)NTOK * IDIM) return;
  size_t row = i / IDIM, c = i % IDIM;
  const float* g = gu + row * (2 * IDIM);
  float gate = g[c], up = g[IDIM + c];
  ff[i] = gate * (1.f / (1.f + __expf(-gate))) * up;
}

// ---------------------------------------------------------------- host
extern "C" void kernel_launch(void* const* d_in, const int* in_sizes, int n_in,
                              void* d_out, int out_size, void* d_ws, size_t ws_size,
                              hipStream_t stream) {
  const int*   ids   = (const int*)d_in[0];
  const int*   pos   = (const int*)d_in[1];
  const float* embed = (const float*)d_in[2];
  const float* Wqkv  = (const float*)d_in[3];
  const float* Wo    = (const float*)d_in[4];
  const float* Wgu   = (const float*)d_in[5];
  const float* Wdn   = (const float*)d_in[6];
  const float* ln1   = (const float*)d_in[7];
  const float* ln2   = (const float*)d_in[8];
  const float* normw = (const float*)d_in[9];
  const float* lmh   = (const float*)d_in[10];
  float* out = (float*)d_out;

  char* p = (char*)d_ws;
  auto carve = [&](size_t bytes) {
    char* r = p;
    p += (bytes + 255) & ~(size_t)255;
    return (float*)r;
  };
  float* h    = carve((size_t)NTOK * HDIM * 4);
  float* xn   = carve((size_t)NTOK * HDIM * 4);
  float* qkv  = carve((size_t)NTOK * QKV_N * 4);
  float* Qb   = carve((size_t)BBATCH * NHEAD * SLEN * HD * 4);
  float* Kb   = carve((size_t)BBATCH * NKVH * SLEN * HD * 4);
  float* Vb   = carve((size_t)BBATCH * NKVH * SLEN * HD * 4);
  float* attn = carve((size_t)NTOK * NHEAD * HD * 4);
  float* gu   = carve((size_t)NTOK * 2 * IDIM * 4);
  float* ff   = carve((size_t)NTOK * IDIM * 4);

  embed_kernel<<<NTOK, 256, 0, stream>>>(ids, embed, h);

  for (int l = 0; l < NLAYER; l++) {
    const float* wqkv = Wqkv + (size_t)l * HDIM * QKV_N;
    const float* wo   = Wo   + (size_t)l * HDIM * HDIM;
    const float* wgu  = Wgu  + (size_t)l * HDIM * (2 * IDIM);
    const float* wdn  = Wdn  + (size_t)l * IDIM * HDIM;

    rmsnorm_kernel<<<NTOK, 256, 0, stream>>>(h, ln1 + (size_t)l * HDIM, xn);
    gemm_kernel<<<dim3(QKV_N / 128, NTOK / 128), 256, 0, stream>>>(
        xn, wqkv, nullptr, qkv, NTOK, QKV_N, HDIM);
    rope_kernel<<<NTOK, 256, 0, stream>>>(qkv, pos, Qb, Kb, Vb);
    attn_kernel<<<dim3(BBATCH * NHEAD, SLEN / QTILE), 128, 0, stream>>>(Qb, Kb, Vb, attn);
    gemm_kernel<<<dim3(HDIM / 128, NTOK / 128), 256, 0, stream>>>(
        attn, wo, h, h, NTOK, HDIM, NHEAD * HD);
    rmsnorm_kernel<<<NTOK, 256, 0, stream>>>(h, ln2 + (size_t)l * HDIM, xn);
    gemm_kernel<<<dim3(2 * IDIM / 128, NTOK / 128), 256, 0, stream>>>(
        xn, wgu, nullptr, gu, NTOK, 2 * IDIM, HDIM);
    size_t nsilu = (size_t)NTOK * IDIM;
    silu_mul_kernel<<<(unsigned)((nsilu + 255) / 256), 256, 0, stream>>>(gu, ff);
    gemm_kernel<<<dim3(HDIM / 128, NTOK / 128), 256, 0, stream>>>(
        ff, wdn, h, h, NTOK, HDIM, IDIM);
  }

  rmsnorm_kernel<<<NTOK, 256, 0, stream>>>(h, normw, xn);
  gemm_kernel<<<dim3(VOCAB / 128, NTOK / 128), 256, 0, stream>>>(
      xn, lmh, nullptr, out, NTOK, VOCAB, HDIM);
}